// IrrepConv1dBlock_197568495890
// MI455X (gfx1250) — compile-verified
//
#include <hip/hip_runtime.h>
#include <hip/hip_bf16.h>
#include <math.h>

// ---------------- constants ----------------
#define LMAXC   3
#define DIR     16          // (LMAX+1)^2
#define CC      128
#define KK      5
#define BB      4
#define NN      64
#define RESG    64          // RES*RES
#define EPSV    1e-5f

typedef __attribute__((ext_vector_type(2))) float v2f;
typedef __attribute__((ext_vector_type(8))) float v8f;
typedef __attribute__((ext_vector_type(4))) unsigned int v4u;
typedef __attribute__((ext_vector_type(8))) int v8i;
typedef __attribute__((ext_vector_type(4))) int v4i;

__device__ __forceinline__ v8f wmma4(v2f a, v2f b, v8f c) {
    // V_WMMA_F32_16X16X4_F32 : D = A(16x4,f32) * B(4x16,f32) + C(16x16,f32)
    return __builtin_amdgcn_wmma_f32_16x16x4_f32(false, a, false, b, (short)0, c,
                                                 false, false);
}

__device__ __forceinline__ float silu(float v) {
    return v / (1.0f + expf(-v));
}

// ---------------------------------------------------------------------------
// Kernel A: compact the 80MB block-diagonal weight into Wc[m][k][co][ci]
// (16*5*128*128 f32 = 5 MiB). Only the diagonal irrep blocks are non-zero.
// ---------------------------------------------------------------------------
__global__ __launch_bounds__(256) void compact_w(const float* __restrict__ w,
                                                 float* __restrict__ Wc) {
    int id = blockIdx.x * 256 + threadIdx.x;          // total 16*5*128*128
    int ci = id & 127;
    int r  = id >> 7;
    int co = r & 127;
    r >>= 7;                                          // r = m*5 + k
    int k = r % 5;
    int m = r / 5;
    Wc[id] = w[((size_t)(co * DIR + m) * (CC * DIR) + (size_t)(ci * DIR + m)) * KK + k];
}

// ---------------------------------------------------------------------------
// Kernel B: conv1d as WMMA GEMM. One workgroup per (m, b):
//   h[co, n] = sum_{k, ci} Wc[m][k][co][ci] * xpad[ci, n+k]  + bias[co*16+m]
//
// The 128x64 f32 activation tile is brought into LDS by the Tensor Data Mover:
// one tensor_load_to_lds descriptor (2-D tile, dim0=64 elems contiguous,
// dim1=128 rows, tensor row stride 1024 elems). TDM padding (4 DWORDs after
// every 64 DWORDs) gives an LDS row stride of 68 floats, and the descriptor's
// LDS base is offset by +2 floats, so the pre-zeroed skip columns are exactly
// the conv's halo padding (cols 0,1,66,67). Issuing wave waits TENSORcnt==0,
// then the workgroup barrier publishes the tile.
// 8 waves; each owns a 16-row c_out strip x all four 16-col n tiles.
// ---------------------------------------------------------------------------
#define XL_STRIDE 68
__global__ __launch_bounds__(256) void conv_wmma(const float* __restrict__ x,
                                                 const float* __restrict__ Wc,
                                                 const float* __restrict__ bias,
                                                 float* __restrict__ hbuf) {
    const int m = blockIdx.x;
    const int b = blockIdx.y;
    __shared__ float XL[CC * XL_STRIDE];     // [ci][npad], npad = n + 2

    const int t = threadIdx.x;
    for (int i = t; i < CC * XL_STRIDE; i += 256) XL[i] = 0.0f;
    __syncthreads();

    if (t < 32) {  // wave 0 issues the TDM descriptor (wave-level op)
        const float* gsrc = x + (size_t)b * (CC * DIR) * NN + (size_t)m * NN;
        unsigned long long ga = (unsigned long long)(uintptr_t)gsrc;
        unsigned int lds = (unsigned int)(uintptr_t)(&XL[0]) + 8u;   // start at col 2
        // D# group0: count=1 (valid), lds_addr, global_addr[56:0], type=2
        v4u g0 = { 1u,
                   lds,
                   (unsigned int)(ga & 0xFFFFFFFFull),
                   (unsigned int)(((ga >> 32) & 0x01FFFFFFull) | 0x80000000u) };
        // D# group1: data_size=4B(code 2), pad_enable, pad_interval=64dw(code 5),
        // pad_amount=4dw(code 3); tensor 64x128, tile 64x128, dim0 stride 1024.
        v8i g1 = { (int)((2u << 16) | (1u << 20) | (5u << 22) | (3u << 25)),
                   (int)(64u << 16),     // tensor_dim0 = 64 (bits 79:48, low half)
                   (int)(128u << 16),    // tensor_dim0 hi = 0 | tensor_dim1 = 128
                   (int)(64u << 16),     // tensor_dim1 hi = 0 | tile_dim0  = 64
                   (int)128,             // tile_dim1 = 128 | tile_dim2 = 0
                   (int)1024,            // tensor_dim0_stride = 1024 (elems)
                   0, 0 };               // stride hi / tensor_dim1_stride = 0
        v4i gz4 = {0, 0, 0, 0};          // 2-D tensor: groups 2/3 unused
        v8i gz8 = {0, 0, 0, 0, 0, 0, 0, 0};
        __builtin_amdgcn_tensor_load_to_lds(g0, g1, gz4, gz4, gz8, 0);
        __builtin_amdgcn_s_wait_tensorcnt(0);
    }
    __syncthreads();

    const int lane   = t & 31;
    const int wave   = t >> 5;
    const int lrow   = lane & 15;       // M / N index within a tile
    const int lk     = (lane >> 4) * 2; // K offset of this lane's fragment pair
    const int cobase = wave * 16;

    v8f acc[4] = {v8f{}, v8f{}, v8f{}, v8f{}};

    const float* wm = Wc + (size_t)m * KK * CC * CC;
    for (int k = 0; k < KK; ++k) {
        const float* wrow = wm + ((size_t)k * CC + cobase + lrow) * CC + lk;
        const int colbase = lrow + k + 2;
        for (int ci0 = 0; ci0 < CC; ci0 += 4) {
            v2f a = *(const v2f*)(wrow + ci0);
            const float* xr0 = &XL[(ci0 + lk) * XL_STRIDE + colbase];
            const float* xr1 = xr0 + XL_STRIDE;
#pragma unroll
            for (int tt = 0; tt < 4; ++tt) {
                v2f bf = { xr0[tt * 16], xr1[tt * 16] };
                acc[tt] = wmma4(a, bf, acc[tt]);
            }
        }
    }

    // epilogue: + bias, scatter into hbuf[row = b*64+n][i = m][c = co]
#pragma unroll
    for (int tt = 0; tt < 4; ++tt) {
        int n = tt * 16 + lrow;
        size_t rowbase = ((size_t)(b * NN + n) * DIR + m) * CC;
#pragma unroll
        for (int j = 0; j < 8; ++j) {
            int co = cobase + j + 8 * (lane >> 4);
            hbuf[rowbase + co] = acc[tt][j] + bias[co * DIR + m];
        }
    }
}

// ---------------------------------------------------------------------------
// Kernel C: fused per-row (row = b*64+n) epilogue.
//   - center l0, irrep-balanced RMS norm, norm_weight/bias
//   - gating: gs = silu(h0 @ gating_w^T + gating_b)            (VALU, tiny)
//   - xg   = to_grid(64x16) @ h(16x128)          -> WMMA, silu
//   - tens = from_grid^T(16x64) @ silu(xg)(64x128) -> WMMA
//   - out[:, i=0] = gs ; out[:, i>=1] = tens
// ---------------------------------------------------------------------------
#define HT_STRIDE 18
#define SL_STRIDE 66
#define FG_STRIDE 66
__global__ __launch_bounds__(256) void fused_post(const float* __restrict__ hbuf,
                                                  const float* __restrict__ nw,
                                                  const float* __restrict__ nb,
                                                  const float* __restrict__ gw,
                                                  const float* __restrict__ gb,
                                                  const float* __restrict__ tg,
                                                  const float* __restrict__ fg,
                                                  float* __restrict__ out) {
    const int row = blockIdx.x;
    const int b = row >> 6;
    const int n = row & 63;

    __shared__ float H[DIR * CC];            // h row, [i][c]
    __shared__ float hT[CC * HT_STRIDE];     // normalized h transposed [c][i]
    __shared__ float FGT[DIR * FG_STRIDE];   // from_grid^T [i][a]
    __shared__ float SlT[CC * SL_STRIDE];    // silu(xg) transposed [c][a]
    __shared__ float red[256];

    const int t = threadIdx.x;
    const float* hr = hbuf + (size_t)row * (DIR * CC);
    for (int i = t; i < DIR * CC; i += 256) H[i] = hr[i];
    for (int i = t; i < RESG * DIR; i += 256) {       // fg is (64,16): [a][i]
        int a = i >> 4, ii = i & 15;
        FGT[ii * FG_STRIDE + a] = fg[i];
    }
    __syncthreads();

    // ---- mean of l0 over c ----
    red[t] = (t < CC) ? H[t] : 0.0f;
    __syncthreads();
    for (int off = 128; off > 0; off >>= 1) {
        if (t < off) red[t] += red[t + off];
        __syncthreads();
    }
    const float mean = red[0] * (1.0f / CC);
    __syncthreads();

    // ---- center l0 + balanced squared-norm partial ----
    float part = 0.0f;
    for (int idx = t; idx < DIR * CC; idx += 256) {
        int i = idx >> 7;
        float v = H[idx];
        if (i == 0) { v -= mean; H[idx] = v; }
        int l = (i == 0) ? 0 : (i < 4) ? 1 : (i < 9) ? 2 : 3;
        float bal = 1.0f / (4.0f * (float)(2 * l + 1));
        part += bal * v * v;
    }
    __syncthreads();
    red[t] = part;
    __syncthreads();
    for (int off = 128; off > 0; off >>= 1) {
        if (t < off) red[t] += red[t + off];
        __syncthreads();
    }
    const float inv = 1.0f / sqrtf(red[0] * (1.0f / CC) + EPSV);
    __syncthreads();

    // ---- normalize, norm_weight/bias, build transposed copy for WMMA B ----
    for (int idx = t; idx < DIR * CC; idx += 256) {
        int i = idx >> 7, c = idx & 127;
        int l = (i == 0) ? 0 : (i < 4) ? 1 : (i < 9) ? 2 : 3;
        float v = H[idx] * inv * nw[l * CC + c];
        if (i == 0) v += nb[c];
        H[idx] = v;
        hT[c * HT_STRIDE + i] = v;
    }
    __syncthreads();

    // ---- gating (uses normalized h0, incl. norm_bias); writes i=0 outputs ----
    if (t < CC) {
        float acc = gb[t];
        const float* g = gw + (size_t)t * CC;
        for (int i = 0; i < CC; ++i) acc += H[i] * g[i];
        out[(size_t)b * (CC * DIR * NN) + (size_t)(t * DIR) * NN + n] = silu(acc);
    }

    const int lane = t & 31;
    const int wave = t >> 5;
    const int lrow = lane & 15;
    const int lk   = (lane >> 4) * 2;

    // ---- GEMM1: xg[a, c] = sum_i to_grid[a,i] * h[i,c]  (M=64, N=128, K=16) ----
    v8f xga[4] = {v8f{}, v8f{}, v8f{}, v8f{}};
    const float* hTb = hT + (wave * 16 + lrow) * HT_STRIDE;
    for (int i0 = 0; i0 < DIR; i0 += 4) {
        v2f bf = *(const v2f*)(hTb + i0 + lk);
#pragma unroll
        for (int tt = 0; tt < 4; ++tt) {
            v2f a = *(const v2f*)(tg + (size_t)(tt * 16 + lrow) * DIR + i0 + lk);
            xga[tt] = wmma4(a, bf, xga[tt]);
        }
    }
    // silu, store transposed [c][a] for GEMM2's B fragments
#pragma unroll
    for (int tt = 0; tt < 4; ++tt) {
#pragma unroll
        for (int j = 0; j < 8; ++j) {
            int a = tt * 16 + j + 8 * (lane >> 4);
            int c = wave * 16 + lrow;
            SlT[c * SL_STRIDE + a] = silu(xga[tt][j]);
        }
    }
    __syncthreads();

    // ---- GEMM2: tens[i, c] = sum_a from_grid[a,i] * silu(xg)[a,c] (M=16,N=128,K=64) ----
    v8f tacc = {};
    for (int a0 = 0; a0 < RESG; a0 += 4) {
        v2f a  = *(const v2f*)(FGT + lrow * FG_STRIDE + a0 + lk);
        v2f bf = *(const v2f*)(SlT + (wave * 16 + lrow) * SL_STRIDE + a0 + lk);
        tacc = wmma4(a, bf, tacc);
    }
#pragma unroll
    for (int j = 0; j < 8; ++j) {
        int i = j + 8 * (lane >> 4);
        if (i > 0) {
            int c = wave * 16 + lrow;
            out[(size_t)b * (CC * DIR * NN) + (size_t)(c * DIR + i) * NN + n] = tacc[j];
        }
    }
}

// ---------------------------------------------------------------------------
extern "C" void kernel_launch(void* const* d_in, const int* in_sizes, int n_in,
                              void* d_out, int out_size, void* d_ws, size_t ws_size,
                              hipStream_t stream) {
    const float* x    = (const float*)d_in[0];
    const float* wgt  = (const float*)d_in[1];
    const float* bias = (const float*)d_in[2];
    const float* nw   = (const float*)d_in[3];
    const float* nb   = (const float*)d_in[4];
    const float* gw   = (const float*)d_in[5];
    const float* gb   = (const float*)d_in[6];
    const float* tg   = (const float*)d_in[7];
    const float* fg   = (const float*)d_in[8];
    float* out = (float*)d_out;

    // workspace: Wc (5 MiB) | hbuf (2 MiB)
    float* Wc   = (float*)d_ws;
    float* hbuf = (float*)((char*)d_ws + (size_t)DIR * KK * CC * CC * sizeof(float));

    // 1) compact block-diagonal weight (16*5*128*128 elements)
    compact_w<<<(DIR * KK * CC * CC) / 256, 256, 0, stream>>>(wgt, Wc);

    // 2) conv as WMMA GEMM (TDM-staged activations), one workgroup per (m, b)
    conv_wmma<<<dim3(DIR, BB), 256, 0, stream>>>(x, Wc, bias, hbuf);

    // 3) fused norm + gating + grid nonlinearity, one workgroup per row
    fused_post<<<BB * NN, 256, 0, stream>>>(hbuf, nw, nb, gw, gb, tg, fg, out);
}